// tE2GN_9912784519270
// MI455X (gfx1250) — compile-verified
//
#include <hip/hip_runtime.h>
#include <math.h>

#define H 8
#define D 28
#define OUTC 30   // D + 2 (wv[28], z, flow_out)

typedef float v2f __attribute__((ext_vector_type(2)));
typedef float v8f __attribute__((ext_vector_type(8)));

// ---------------------------------------------------------------------------
// Pass 0: zero the output buffer (harness poisons it with 0xAA).
// ---------------------------------------------------------------------------
__global__ void tE2GN_zero_kernel(float4* __restrict__ p, long n4, float* __restrict__ ptail,
                                  long ntail) {
  long i = (long)blockIdx.x * blockDim.x + threadIdx.x;
  long stride = (long)gridDim.x * blockDim.x;
  for (; i < n4; i += stride) p[i] = make_float4(0.f, 0.f, 0.f, 0.f);
  if (blockIdx.x == 0 && threadIdx.x < ntail) ptail[threadIdx.x] = 0.f;
}

// ---------------------------------------------------------------------------
// Pass 1: one wave32 handles 2 edges (16 (edge,head) rows).
//   A row r (r=0..15)  = k[src[e0 + r/8], head r%8, :]   (16 x 28, f32)
//   B col r            = q[dst[e0 + r/8], head r%8, :]   (28 x 16, f32)
//   diag(A x B) = raw attention scores, via 7 chained V_WMMA_F32_16X16X4_F32
//   (full f32 precision, K = 7*4 = 28 exactly, no padding).
//
// f32 16x16x4 operand layout (ISA 7.12.2):
//   A: lane l<16 -> row l,  V0=K0, V1=K1 ; lane l>=16 -> row l-16, V0=K2, V1=K3
//   B: lane l<16 -> col l,  V0=K0, V1=K1 ; lane l>=16 -> col l-16, V0=K2, V1=K3
//   C/D: VGPR r, lanes 0-15 -> D[r, lane]; lanes 16-31 -> D[r+8, lane-16]
//   => diag D[i,i]: lanes 0-7 hold it in acc[lane]; lanes 24-31 in acc[lane-24]
//
// Out-of-range rows (last partial wave) use clamped indices: their fragment
// data is garbage but D[i,i] only touches row i of A / col i of B, so valid
// rows' diagonals are unaffected; invalid rows' scores are discarded.
// All fragment loads are therefore unconditional, 8-byte aligned float2s.
// ---------------------------------------------------------------------------
__global__ __launch_bounds__(256) void tE2GN_pass1_kernel(
    const float* __restrict__ k, const float* __restrict__ q, const float* __restrict__ v,
    const int* __restrict__ src, const int* __restrict__ dst,
    float* __restrict__ out, float* __restrict__ ws_score, int E, float inv_scale) {
  const int lane = threadIdx.x & 31;
  const int wib  = threadIdx.x >> 5;                       // wave index in block (0..7)
  const long wave = (long)blockIdx.x * (blockDim.x >> 5) + wib;
  const long e0 = wave * 2;

  __shared__ float s_score[8][16];
  __shared__ int   s_srcn[8][16];
  __shared__ int   s_dstn[8][16];

  // (edge,head) row served by this lane for the fragments
  const int  r    = lane & 15;
  const long er   = e0 + (r >> 3);
  const long erc  = (er < (long)E) ? er : 0;               // clamp for safe loads
  const int  head = r & 7;
  const int  sn   = src[erc];
  const int  dn   = dst[erc];
  const float* kbase = k + (long)sn * (H * D) + head * D;  // A row r
  const float* qbase = q + (long)dn * (H * D) + head * D;  // B col r
  const int koff = (lane < 16) ? 0 : 2;

  // Hoist all 14 float2 (b64) loads ahead of the WMMA chain.
  v2f a[7], b[7];
#pragma unroll
  for (int c = 0; c < 7; ++c) a[c] = *(const v2f*)(kbase + 4 * c + koff);
#pragma unroll
  for (int c = 0; c < 7; ++c) b[c] = *(const v2f*)(qbase + 4 * c + koff);

  v8f acc = {0.f, 0.f, 0.f, 0.f, 0.f, 0.f, 0.f, 0.f};
#pragma unroll
  for (int c = 0; c < 7; ++c) {
    // 8 args: (neg_a, A, neg_b, B, c_mod, C, reuse_a, reuse_b)
    acc = __builtin_amdgcn_wmma_f32_16x16x4_f32(false, a[c], false, b[c], (short)0, acc,
                                                false, false);
  }

  // Extract the diagonal this lane owns (if any).
  const bool isDiag = (lane < 8) || (lane >= 24);
  const int  dr = (lane < 8) ? lane : (lane - 16);         // diag row index 0..15
  float raw = 0.f;
#pragma unroll
  for (int j = 0; j < 8; ++j)
    if ((lane & 7) == j) raw = acc[j];

  if (isDiag) {
    const long ed = e0 + (dr >> 3);
    const int  hd = dr & 7;
    float sc = 0.f;
    int dsn = 0, ddn = 0;
    if (ed < (long)E) {
      dsn = src[ed];
      ddn = dst[ed];
      float x = raw * inv_scale;
      x  = fminf(fmaxf(x, -5.f), 5.f);
      sc = __expf(x);
      atomicAdd(out + (long)ddn * (H * OUTC) + hd * OUTC + D, sc);   // z
      if (ws_score) ws_score[ed * H + hd] = sc;                      // cache for pass 2
    }
    s_score[wib][dr] = sc;
    s_srcn[wib][dr]  = dsn;
    s_dstn[wib][dr]  = ddn;
  }
  __syncthreads();

  // wv accumulation: 16 rows x 28 dims = 448 atomics, spread over 32 lanes.
#pragma unroll 1
  for (int idx = lane; idx < 16 * D; idx += 32) {
    const int rr = idx / D;
    const int dd = idx - rr * D;
    const float sc = s_score[wib][rr];
    if (sc != 0.f) {
      const int hh = rr & 7;
      const float val = v[(long)s_srcn[wib][rr] * (H * D) + hh * D + dd] * sc;
      atomicAdd(out + (long)s_dstn[wib][rr] * (H * OUTC) + hh * OUTC + dd, val);
    }
  }
}

// ---------------------------------------------------------------------------
// Pass 2: one thread per (edge, head):
//   flow_out[src,h] += flow_score[dst,h] * score / z[dst,h]
// score comes from the workspace when it fit, else recomputed (f32 dot).
// ---------------------------------------------------------------------------
__global__ __launch_bounds__(256) void tE2GN_pass2_kernel(
    const float* __restrict__ k, const float* __restrict__ q,
    const float* __restrict__ flow_score,
    const int* __restrict__ src, const int* __restrict__ dst,
    const float* __restrict__ ws_score, float* __restrict__ out, long EH, float inv_scale) {
  long t = (long)blockIdx.x * blockDim.x + threadIdx.x;
  if (t >= EH) return;
  const long e = t >> 3;
  const int  h = (int)(t & 7);
  const int sn = src[e];
  const int dn = dst[e];
  float sc;
  if (ws_score) {
    sc = ws_score[t];
  } else {
    const float* kb = k + (long)sn * (H * D) + h * D;
    const float* qb = q + (long)dn * (H * D) + h * D;
    float a = 0.f;
#pragma unroll
    for (int dd = 0; dd < D; ++dd) a = fmaf(kb[dd], qb[dd], a);
    a  = a * inv_scale;
    a  = fminf(fmaxf(a, -5.f), 5.f);
    sc = __expf(a);
  }
  const float z  = out[(long)dn * (H * OUTC) + h * OUTC + D];
  const float fo = flow_score[(long)dn * H + h] * sc / z;
  atomicAdd(out + (long)sn * (H * OUTC) + h * OUTC + (D + 1), fo);
}

// ---------------------------------------------------------------------------
extern "C" void kernel_launch(void* const* d_in, const int* in_sizes, int n_in,
                              void* d_out, int out_size, void* d_ws, size_t ws_size,
                              hipStream_t stream) {
  const float* k   = (const float*)d_in[0];
  const float* q   = (const float*)d_in[1];
  const float* v   = (const float*)d_in[2];
  const float* fsc = (const float*)d_in[3];
  const int*   src = (const int*)d_in[4];
  const int*   dst = (const int*)d_in[5];
  float* out = (float*)d_out;

  const int  E = in_sizes[4];
  const float inv_scale = 1.0f / sqrtf((float)D);

  // Pass 0: zero d_out (out_size = N*H*30, divisible by 4, but handle tail anyway).
  const long n4    = (long)out_size / 4;
  const long ntail = (long)out_size - n4 * 4;
  {
    long blocks = (n4 + 255) / 256;
    if (blocks < 1) blocks = 1;
    if (blocks > 65535) blocks = 65535;  // grid-stride covers the rest
    tE2GN_zero_kernel<<<(int)blocks, 256, 0, stream>>>((float4*)out, n4,
                                                       out + n4 * 4, ntail);
  }

  // Use workspace for per-(edge,head) scores if it fits (E*H*4 = 32 MB here).
  float* ws_score =
      (ws_size >= (size_t)E * H * sizeof(float)) ? (float*)d_ws : nullptr;

  // Pass 1: 2 edges per wave, 8 waves (256 threads) per block.
  const long waves   = ((long)E + 1) / 2;
  const long blocks1 = (waves + 7) / 8;
  tE2GN_pass1_kernel<<<(int)blocks1, 256, 0, stream>>>(k, q, v, src, dst, out,
                                                       ws_score, E, inv_scale);

  // Pass 2: one thread per (edge, head).
  const long EH      = (long)E * H;
  const long blocks2 = (EH + 255) / 256;
  tE2GN_pass2_kernel<<<(int)blocks2, 256, 0, stream>>>(k, q, fsc, src, dst,
                                                       ws_score, out, EH, inv_scale);
}